// NetGAT_32504312496301
// MI455X (gfx1250) — compile-verified
//
#include <hip/hip_runtime.h>
#include <hip/hip_bf16.h>
#include <math.h>
#include <limits.h>

typedef float v2f __attribute__((ext_vector_type(2)));
typedef float v8f __attribute__((ext_vector_type(8)));

#define BT 256

// ---------------------------------------------------------------------------
// Transpose W[K,Cout] -> Wt[Cout,K] so B fragments load as contiguous b64.
// ---------------------------------------------------------------------------
__global__ void transpose_w(const float* __restrict__ W, float* __restrict__ Wt,
                            int K, int Cout) {
  int t = blockIdx.x * blockDim.x + threadIdx.x;
  if (t >= K * Cout) return;
  int k = t / Cout, c = t - k * Cout;
  Wt[(size_t)c * K + k] = W[t];
}

// ---------------------------------------------------------------------------
// GEMM: Out[N,Cout] = X[N,K] @ W[K,Cout]  (W pre-transposed: Wt[Cout,K]).
// One wave computes a 16 x (16*CT) strip: the A fragment (16x4 f32) is loaded
// once per k-step and fed to CT back-to-back V_WMMA_F32_16X16X4_F32 ops.
// K, Cout compile-time -> all strides are immediates (no v_mul_u64 addressing).
// A layout: lane&15 = row M; lane>>4 selects K pair {0,1} vs {2,3}.
// B layout: lane&15 = col N; lane>>4 selects K pair {0,1} vs {2,3}.
// D layout: VGPR r -> row (r + 8*(lane>>4)), col (lane&15).
// ---------------------------------------------------------------------------
template <int K, int Cout, int CT>
__global__ void __launch_bounds__(BT)
gemm_wmma(const float* __restrict__ X, const float* __restrict__ Wt,
          float* __restrict__ Out, int N) {
  constexpr int groupsX = Cout / (16 * CT);
  const int gid  = blockIdx.x * blockDim.x + threadIdx.x;
  const int wave = gid >> 5;
  const int lane = gid & 31;
  const int tilesY = N >> 4;
  if (wave >= groupsX * tilesY) return;
  const int tr = wave / groupsX;
  const int tg = wave - tr * groupsX;
  const int rowBase = tr << 4;
  const int colBase = tg * (16 * CT);
  const int half = lane >> 4, l = lane & 15;

  v8f acc[CT];
#pragma unroll
  for (int c = 0; c < CT; ++c) acc[c] = (v8f){};

#if __has_builtin(__builtin_amdgcn_wmma_f32_16x16x4_f32)
  // per-lane streaming pointers; strides are compile-time constants
  const float* __restrict__ xr = X + (size_t)(rowBase + l) * K + (half << 1);
  const float* __restrict__ wt = Wt + (size_t)(colBase + l) * K + (half << 1);
#pragma unroll 4
  for (int k = 0; k < K; k += 4) {
    v2f a;
    a.x = xr[0];
    a.y = xr[1];
#pragma unroll
    for (int c = 0; c < CT; ++c) {
      v2f b;
      b.x = wt[c * 16 * K + 0];   // contiguous pair in Wt -> b64 load
      b.y = wt[c * 16 * K + 1];
      acc[c] = __builtin_amdgcn_wmma_f32_16x16x4_f32(false, a, false, b,
                                                     (short)0, acc[c],
                                                     false, false);
    }
    xr += 4;
    wt += 4;
  }
#else
  // scalar fallback with identical D-layout semantics (keeps file compiling)
  for (int c = 0; c < CT; ++c)
    for (int r = 0; r < 8; ++r) {
      float s = 0.f;
      const float* xr2 = X + (size_t)(rowBase + r + (half << 3)) * K;
      const float* wt2 = Wt + (size_t)(colBase + c * 16 + l) * K;
      for (int k = 0; k < K; ++k) s += xr2[k] * wt2[k];
      acc[c][r] = s;
    }
#endif

#pragma unroll
  for (int c = 0; c < CT; ++c) {
    float* orow = Out + (size_t)(rowBase + (half << 3)) * Cout + colBase + c * 16 + l;
#pragma unroll
    for (int r = 0; r < 8; ++r) orow[(size_t)r * Cout] = acc[c][r];
  }
}

// ---------------------------------------------------------------------------
// Attention scores: es[n,h] = <h[n,h,:], a_s[h,:]>, ed likewise.
// ---------------------------------------------------------------------------
__global__ void attn_scores(const float* __restrict__ Hm,
                            const float* __restrict__ As,
                            const float* __restrict__ Ad,
                            float* __restrict__ es, float* __restrict__ ed,
                            int N, int H, int C) {
  int t = blockIdx.x * blockDim.x + threadIdx.x;
  if (t >= N * H) return;
  int n = t / H, h = t - n * H;
  const float* hp = Hm + (size_t)n * H * C + (size_t)h * C;
  const float* as = As + (size_t)h * C;
  const float* ad = Ad + (size_t)h * C;
  float s = 0.f, d = 0.f;
  for (int c = 0; c < C; ++c) {
    float v = hp[c];
    s = fmaf(v, as[c], s);
    d = fmaf(v, ad[c], d);
  }
  es[t] = s;
  ed[t] = d;
}

__global__ void fill_f32(float* __restrict__ p, float v, int n) {
  int t = blockIdx.x * blockDim.x + threadIdx.x;
  if (t < n) p[t] = v;
}
__global__ void fill_i32(int* __restrict__ p, int v, int n) {
  int t = blockIdx.x * blockDim.x + threadIdx.x;
  if (t < n) p[t] = v;
}

// ordered-int mapping so atomicMax(int) orders floats correctly
__device__ __forceinline__ int f2o(float f) {
  int i = __float_as_int(f);
  return i >= 0 ? i : (i ^ 0x7fffffff);
}
__device__ __forceinline__ float o2f(int o) {
  int i = o >= 0 ? o : (o ^ 0x7fffffff);
  return __int_as_float(i);
}

__device__ __forceinline__ void edge_endpoints(const long long* __restrict__ ei,
                                               int E, int eidx, int& s, int& d) {
  if (eidx < E) {
    s = (int)ei[eidx];
    d = (int)ei[(size_t)E + eidx];
  } else {
    s = d = eidx - E;  // self-loop
  }
}

__device__ __forceinline__ float lrelu(float x) {
  return x > 0.f ? x : 0.2f * x;
}

// pass 1: per-dest segment max of leaky_relu(es[src]+ed[dst])
__global__ void edge_max(const long long* __restrict__ ei, int E, int N,
                         const float* __restrict__ es,
                         const float* __restrict__ ed,
                         int* __restrict__ m, int H) {
  int t = blockIdx.x * blockDim.x + threadIdx.x;
  if (t >= E + N) return;
  int s, d;
  edge_endpoints(ei, E, t, s, d);
  for (int h = 0; h < H; ++h) {
    float e = lrelu(es[(size_t)s * H + h] + ed[(size_t)d * H + h]);
    atomicMax(&m[(size_t)d * H + h], f2o(e));
  }
}

// pass 2: denom[dst,h] += exp(e - m[dst,h])
__global__ void edge_sum(const long long* __restrict__ ei, int E, int N,
                         const float* __restrict__ es,
                         const float* __restrict__ ed,
                         const int* __restrict__ m,
                         float* __restrict__ den, int H) {
  int t = blockIdx.x * blockDim.x + threadIdx.x;
  if (t >= E + N) return;
  int s, d;
  edge_endpoints(ei, E, t, s, d);
  for (int h = 0; h < H; ++h) {
    float e = lrelu(es[(size_t)s * H + h] + ed[(size_t)d * H + h]);
    float w = expf(e - o2f(m[(size_t)d * H + h]));
    atomicAdd(&den[(size_t)d * H + h], w);
  }
}

// pass 3: Out[dst,h,:] += h[src,h,:] * alpha  (thread per (edge, head));
// source row gathered as float4 (rows are 64B aligned), L2-resident atomics out.
__global__ void edge_agg(const long long* __restrict__ ei, int E, int N,
                         const float* __restrict__ es,
                         const float* __restrict__ ed,
                         const int* __restrict__ m,
                         const float* __restrict__ den,
                         const float* __restrict__ Hm,
                         float* __restrict__ Out, int H, int C) {
  int t = blockIdx.x * blockDim.x + threadIdx.x;
  int ET = E + N;
  if (t >= ET * H) return;
  int eidx = t / H, h = t - eidx * H;
  int s, d;
  edge_endpoints(ei, E, eidx, s, d);
  float e = lrelu(es[(size_t)s * H + h] + ed[(size_t)d * H + h]);
  float alpha = expf(e - o2f(m[(size_t)d * H + h])) / den[(size_t)d * H + h];
  const float4* __restrict__ hp4 =
      reinterpret_cast<const float4*>(Hm + ((size_t)s * H + h) * C);
  float* op = Out + ((size_t)d * H + h) * C;
  for (int q = 0; q < (C >> 2); ++q) {
    float4 v = hp4[q];
    atomicAdd(&op[q * 4 + 0], v.x * alpha);
    atomicAdd(&op[q * 4 + 1], v.y * alpha);
    atomicAdd(&op[q * 4 + 2], v.z * alpha);
    atomicAdd(&op[q * 4 + 3], v.w * alpha);
  }
}

__global__ void bias_act(float* __restrict__ X, const float* __restrict__ b,
                         int N, int D, int relu) {
  int t = blockIdx.x * blockDim.x + threadIdx.x;
  if (t >= N * D) return;
  int c = t % D;
  float v = X[t] + b[c];
  if (relu) v = v > 0.f ? v : 0.f;
  X[t] = v;
}

__global__ void final_logsoftmax(const float* __restrict__ Agg,
                                 const float* __restrict__ b,
                                 float* __restrict__ Out, int N) {
  int n = blockIdx.x * blockDim.x + threadIdx.x;
  if (n >= N) return;
  float v[32];
  float mx = -INFINITY;
#pragma unroll
  for (int c = 0; c < 32; ++c) {
    v[c] = Agg[(size_t)n * 32 + c] + b[c];
    mx = fmaxf(mx, v[c]);
  }
  float sum = 0.f;
#pragma unroll
  for (int c = 0; c < 32; ++c) sum += expf(v[c] - mx);
  float lse = mx + logf(sum);
#pragma unroll
  for (int c = 0; c < 32; ++c) Out[(size_t)n * 32 + c] = v[c] - lse;
}

// ---------------------------------------------------------------------------

static inline int cdiv_ll(long long a, long long b) { return (int)((a + b - 1) / b); }

extern "C" void kernel_launch(void* const* d_in, const int* in_sizes, int n_in,
                              void* d_out, int out_size, void* d_ws, size_t ws_size,
                              hipStream_t stream) {
  const float*     x   = (const float*)d_in[0];
  const long long* ei  = (const long long*)d_in[1];   // int64 edge_index [2,E]
  const float*     W1  = (const float*)d_in[2];
  const float*     a1s = (const float*)d_in[3];
  const float*     a1d = (const float*)d_in[4];
  const float*     b1  = (const float*)d_in[5];
  const float*     W2  = (const float*)d_in[6];
  const float*     a2s = (const float*)d_in[7];
  const float*     a2d = (const float*)d_in[8];
  const float*     b2  = (const float*)d_in[9];
  const float*     W3  = (const float*)d_in[10];
  const float*     a3s = (const float*)d_in[11];
  const float*     a3d = (const float*)d_in[12];
  const float*     b3  = (const float*)d_in[13];

  constexpr int nfeat = 128, nhid = 16, nclass = 32, H = 15;
  constexpr int d1 = H * nhid;           // 240
  const int N  = in_sizes[0] / nfeat;    // 30000
  const int E  = in_sizes[1] / 2;        // 600000
  const int ET = E + N;                  // edges incl. self-loops

  // workspace carve-up (floats)
  float* ws   = (float*)d_ws;
  float* bufA = ws;                              // N*d1  : h of current layer
  float* bufB = bufA + (size_t)N * d1;           // N*d1  : aggregation / next x
  float* es   = bufB + (size_t)N * d1;           // N*H
  float* ed   = es   + (size_t)N * H;            // N*H
  int*   m    = (int*)(ed + (size_t)N * H);      // N*H
  float* den  = (float*)(m + (size_t)N * H);     // N*H
  float* wt1  = den + (size_t)N * H;             // d1*nfeat
  float* wt2  = wt1 + (size_t)d1 * nfeat;        // d1*d1
  float* wt3  = wt2 + (size_t)d1 * d1;           // nclass*d1

  const int NEG = (int)0x80000000;  // below any ordered-float encoding

  // pre-transpose weights (tiny; B fragments then load as contiguous b64)
  transpose_w<<<cdiv_ll((long long)nfeat * d1, BT), BT, 0, stream>>>(W1, wt1, nfeat, d1);
  transpose_w<<<cdiv_ll((long long)d1 * d1, BT), BT, 0, stream>>>(W2, wt2, d1, d1);
  transpose_w<<<cdiv_ll((long long)d1 * nclass, BT), BT, 0, stream>>>(W3, wt3, d1, nclass);

  // ---------------- layer 1 (H=15, C=16) ----------------
  {
    long long waves = (long long)(N / 16) * (d1 / 80);  // CT=5 -> 3 groups/row
    gemm_wmma<nfeat, d1, 5><<<cdiv_ll(waves * 32, BT), BT, 0, stream>>>(x, wt1, bufA, N);
  }
  attn_scores<<<cdiv_ll((long long)N * H, BT), BT, 0, stream>>>(bufA, a1s, a1d, es, ed, N, H, nhid);
  fill_i32<<<cdiv_ll((long long)N * H, BT), BT, 0, stream>>>(m, NEG, N * H);
  fill_f32<<<cdiv_ll((long long)N * H, BT), BT, 0, stream>>>(den, 0.f, N * H);
  fill_f32<<<cdiv_ll((long long)N * d1, BT), BT, 0, stream>>>(bufB, 0.f, N * d1);
  edge_max<<<cdiv_ll(ET, BT), BT, 0, stream>>>(ei, E, N, es, ed, m, H);
  edge_sum<<<cdiv_ll(ET, BT), BT, 0, stream>>>(ei, E, N, es, ed, m, den, H);
  edge_agg<<<cdiv_ll((long long)ET * H, BT), BT, 0, stream>>>(ei, E, N, es, ed, m, den, bufA, bufB, H, nhid);
  bias_act<<<cdiv_ll((long long)N * d1, BT), BT, 0, stream>>>(bufB, b1, N, d1, 1);

  // ---------------- layer 2 (H=15, C=16) ----------------
  {
    long long waves = (long long)(N / 16) * (d1 / 80);
    gemm_wmma<d1, d1, 5><<<cdiv_ll(waves * 32, BT), BT, 0, stream>>>(bufB, wt2, bufA, N);
  }
  attn_scores<<<cdiv_ll((long long)N * H, BT), BT, 0, stream>>>(bufA, a2s, a2d, es, ed, N, H, nhid);
  fill_i32<<<cdiv_ll((long long)N * H, BT), BT, 0, stream>>>(m, NEG, N * H);
  fill_f32<<<cdiv_ll((long long)N * H, BT), BT, 0, stream>>>(den, 0.f, N * H);
  fill_f32<<<cdiv_ll((long long)N * d1, BT), BT, 0, stream>>>(bufB, 0.f, N * d1);
  edge_max<<<cdiv_ll(ET, BT), BT, 0, stream>>>(ei, E, N, es, ed, m, H);
  edge_sum<<<cdiv_ll(ET, BT), BT, 0, stream>>>(ei, E, N, es, ed, m, den, H);
  edge_agg<<<cdiv_ll((long long)ET * H, BT), BT, 0, stream>>>(ei, E, N, es, ed, m, den, bufA, bufB, H, nhid);
  bias_act<<<cdiv_ll((long long)N * d1, BT), BT, 0, stream>>>(bufB, b2, N, d1, 1);

  // ---------------- layer 3 (H=1, C=32) ----------------
  {
    long long waves = (long long)(N / 16) * 1;          // CT=2 -> 1 group/row
    gemm_wmma<d1, nclass, 2><<<cdiv_ll(waves * 32, BT), BT, 0, stream>>>(bufB, wt3, bufA, N);
  }
  attn_scores<<<cdiv_ll(N, BT), BT, 0, stream>>>(bufA, a3s, a3d, es, ed, N, 1, nclass);
  fill_i32<<<cdiv_ll(N, BT), BT, 0, stream>>>(m, NEG, N);
  fill_f32<<<cdiv_ll(N, BT), BT, 0, stream>>>(den, 0.f, N);
  fill_f32<<<cdiv_ll((long long)N * nclass, BT), BT, 0, stream>>>(bufB, 0.f, N * nclass);
  edge_max<<<cdiv_ll(ET, BT), BT, 0, stream>>>(ei, E, N, es, ed, m, 1);
  edge_sum<<<cdiv_ll(ET, BT), BT, 0, stream>>>(ei, E, N, es, ed, m, den, 1);
  edge_agg<<<cdiv_ll(ET, BT), BT, 0, stream>>>(ei, E, N, es, ed, m, den, bufA, bufB, 1, nclass);

  final_logsoftmax<<<cdiv_ll(N, BT), BT, 0, stream>>>(bufB, b3, (float*)d_out, N);
}